// NeRF_75359496176026
// MI455X (gfx1250) — compile-verified
//
#include <hip/hip_runtime.h>
#include <cstdint>
#include <cstddef>

// ---------------------------------------------------------------------------
// NeRF volume-rendering composite for MI455X (gfx1250).
//   out[n,c] = (sum_s ts[s]*alpha[s]) * (sum_s sigmoid(rgbo[n,s,c]))
// with ts[s] = exp(cumprod(mult)[s-1]), ts[0]=1, mult = -opacity*delta.
//
// Memory-bound (~168 MB @ 23.3 TB/s => ~7 us floor). One wave32 per ray;
// ray tiles are staged Global->LDS with CDNA5 async-to-LDS DMA (ASYNCcnt),
// double-buffered so the next ray streams in while the current one computes.
// The sequential cumprod is a wave32 Hillis-Steele prefix *product* (5
// shuffle steps per 32-sample chunk) chained across 4 chunks.
// ---------------------------------------------------------------------------

#define FAR_DELTA_F 1.0e9f

constexpr int S      = 128;          // samples per ray
constexpr int WV     = 8;            // waves per block (256 threads)
constexpr int RGBO_F = S * 4;        // 512 floats (2048 B) per ray
constexpr int BUF_F  = RGBO_F + S;   // + 128 depth floats = 2560 B per buffer

__device__ __forceinline__ float fast_exp(float x) {
  return __builtin_amdgcn_exp2f(x * 1.44269504088896340736f);  // v_exp_f32
}
__device__ __forceinline__ float fast_sigmoid(float x) {
  return __builtin_amdgcn_rcpf(1.0f + fast_exp(-x));           // v_rcp_f32
}

__global__ __launch_bounds__(WV * 32, 1)
void nerf_composite_kernel(const float* __restrict__ rgbo,
                           const float* __restrict__ depth,
                           float* __restrict__ out,
                           int nRays)
{
  // Per-wave private double buffer: [rgbo 2048B][depth 512B] x2, 8 waves.
  __shared__ __align__(16) float lds[WV * 2 * BUF_F];

  const int lane = threadIdx.x & 31;
  const int wib  = threadIdx.x >> 5;
  const int wave = blockIdx.x * WV + wib;
  const int nWav = gridDim.x * WV;

  float* wbase = &lds[wib * 2 * BUF_F];
  // Low 32 bits of the flat shared address == LDS byte offset (aperture in
  // the high bits), which is what the async-to-LDS VDST VGPR wants.
  const uint32_t ldsBase = (uint32_t)(uintptr_t)wbase;

  // ---- async stage: copy one ray's tile (2048B rgbo + 512B depth) to LDS.
  // Each lane moves 16B per op => 512B per wave per instruction; 5 ops/ray.
  auto stage = [&](int ray, int buf) {
    const uint64_t gR =
        (uint64_t)(uintptr_t)(rgbo + (size_t)ray * RGBO_F) + (uint64_t)(lane * 16);
    const uint32_t lB =
        ldsBase + (uint32_t)(buf * (BUF_F * 4)) + (uint32_t)(lane * 16);
#pragma unroll
    for (int j = 0; j < 4; ++j) {
      uint32_t la = lB + (uint32_t)(j * 512);
      uint64_t ga = gR + (uint64_t)(j * 512);
      asm volatile("global_load_async_to_lds_b128 %0, %1, off"
                   :: "v"(la), "v"(ga) : "memory");
    }
    const uint64_t gD =
        (uint64_t)(uintptr_t)(depth + (size_t)ray * S) + (uint64_t)(lane * 16);
    const uint32_t lD = lB + (uint32_t)(RGBO_F * 4);
    asm volatile("global_load_async_to_lds_b128 %0, %1, off"
                 :: "v"(lD), "v"(gD) : "memory");
  };

  // ---- compute one ray from its LDS buffer.
  auto compute = [&](int ray, int buf) {
    const float4* rg = (const float4*)(wbase + buf * BUF_F);
    const float*  dp = wbase + buf * BUF_F + RGBO_F;

    float R = 1.0f;            // running product of mult over previous chunks
    float wsum = 0.0f, sr = 0.0f, sg = 0.0f, sb = 0.0f;

#pragma unroll
    for (int j = 0; j < 4; ++j) {
      const int s = j * 32 + lane;
      const float4 v = rg[s];
      const float  d = dp[s];
      const float  dn = dp[(s < S - 1) ? (s + 1) : s];
      const float  delta = (s == S - 1) ? FAR_DELTA_F : (dn - d);
      const float  m = -v.w * delta;

      // Wave32 inclusive prefix product of m (Hillis-Steele, 5 steps).
      float ip = m;
#pragma unroll
      for (int off = 1; off < 32; off <<= 1) {
        float t = __shfl_up(ip, off, 32);
        if (lane >= off) ip *= t;
      }
      float prev = __shfl_up(ip, 1, 32);
      float excl = (lane == 0) ? 1.0f : prev;     // product of m before s (chunk)
      float c    = R * excl;                       // global cumprod up to s-1
      float ts   = (s == 0) ? 1.0f : fast_exp(c);  // exp(0)=1 special case
      float alpha = 1.0f - fast_exp(m);
      wsum += ts * alpha;

      sr += fast_sigmoid(v.x);
      sg += fast_sigmoid(v.y);
      sb += fast_sigmoid(v.z);

      R *= __shfl(ip, 31, 32);                     // carry chunk product
    }

    // Full-wave butterfly reductions (all lanes end with the totals).
#pragma unroll
    for (int off = 16; off >= 1; off >>= 1) {
      wsum += __shfl_xor(wsum, off, 32);
      sr   += __shfl_xor(sr,   off, 32);
      sg   += __shfl_xor(sg,   off, 32);
      sb   += __shfl_xor(sb,   off, 32);
    }
    if (lane < 3) {
      float ch = (lane == 0) ? sr : (lane == 1) ? sg : sb;
      out[(size_t)ray * 3 + lane] = wsum * ch;
    }
  };

  // ---- grid-stride loop with double-buffered async pipeline.
  int ray = wave;
  if (ray < nRays) stage(ray, 0);
  int buf = 0;
  for (; ray < nRays; ray += nWav) {
    const int  nxt     = ray + nWav;
    const bool hasNext = nxt < nRays;
    if (hasNext) {
      stage(nxt, buf ^ 1);
      // 10 async ops in flight; loads complete in order, so <=5 outstanding
      // means the *current* buffer's 5 transfers have landed in LDS.
      asm volatile("s_wait_asynccnt 0x5" ::: "memory");
    } else {
      asm volatile("s_wait_asynccnt 0x0" ::: "memory");
    }
    compute(ray, buf);
    buf ^= 1;
  }
}

extern "C" void kernel_launch(void* const* d_in, const int* in_sizes, int n_in,
                              void* d_out, int out_size, void* d_ws, size_t ws_size,
                              hipStream_t stream) {
  (void)n_in; (void)d_ws; (void)ws_size; (void)out_size;
  const float* rgbo  = (const float*)d_in[0];   // [N, S, 4] f32
  const float* depth = (const float*)d_in[1];   // [N, S]    f32
  float* out = (float*)d_out;                   // [N, 3]    f32

  const int nRays = in_sizes[0] / (S * 4);
  int blocks = (nRays + WV - 1) / WV;
  if (blocks > 2048) blocks = 2048;             // ~16K waves, 4 rays/wave
  if (blocks < 1) blocks = 1;
  nerf_composite_kernel<<<blocks, WV * 32, 0, stream>>>(rgbo, depth, out, nRays);
}